// Sequence_59554016526399
// MI455X (gfx1250) — compile-verified
//
#include <hip/hip_runtime.h>
#include <math.h>

// ---------------------------------------------------------------------------
// CDNA5 WMMA f32 16x16x4:  A[16,4] (2 VGPRs), B[4,16] (2 VGPRs),
// C/D[16,16] (8 VGPRs).  Per ISA layouts:
//   A: lane<16 -> M=lane, {K=0,K=1}; lane>=16 -> M=lane-16, {K=2,K=3}
//   B: lane<16 -> N=lane, rows {K=0,K=1}; lane>=16 -> rows {K=2,K=3}
//   C/D: vgpr r -> row (r + 8*(lane>=16)), col = lane&15
// ---------------------------------------------------------------------------
typedef __attribute__((ext_vector_type(2))) float v2f;
typedef __attribute__((ext_vector_type(8))) float v8f;

__device__ __forceinline__ v8f wmma_16x16x4(v2f a, v2f b, v8f c) {
  // 8 args: (neg_a, A, neg_b, B, c_mod, C, reuse_a, reuse_b)
  return __builtin_amdgcn_wmma_f32_16x16x4_f32(false, a, false, b, (short)0, c,
                                               false, false);
}

__device__ __forceinline__ float fast_sigmoid(float x) {
  float e = __expf(-x);
  return __builtin_amdgcn_rcpf(1.0f + e);
}
__device__ __forceinline__ float fast_tanh(float x) {
  // tanh(x) = 2/(1+e^{-2x}) - 1
  float u = __expf(-2.0f * x);
  return fmaf(2.0f, __builtin_amdgcn_rcpf(1.0f + u), -1.0f);
}

// Async global->LDS staging (GLOBAL_LOAD_ASYNC_TO_LDS_B128 / ASYNCcnt),
// guarded so the file still compiles on toolchains without the builtins.
// Probe-confirmed signature (from the round-2 diagnostic): the b128 form
// takes (int4 AS1*, int4 AS3*, imm offset, imm cpol) with GCC-vector int4.
#if defined(__gfx1250__) &&                                          \
    __has_builtin(__builtin_amdgcn_global_load_async_to_lds_b128) && \
    __has_builtin(__builtin_amdgcn_s_wait_asynccnt)
#define USE_ASYNC_STAGE 1
typedef int v4i_g __attribute__((vector_size(16)));
typedef __attribute__((address_space(1))) v4i_g as1_v4i;
typedef __attribute__((address_space(3))) v4i_g as3_v4i;
#else
#define USE_ASYNC_STAGE 0
#endif

// ---------------------------------------------------------------------------
// Layer-1 input projection (both directions), fused embedding gather.
// xg[t][q*32 + l] = sum_k emb[tok[t']][k] * Wih[q*30+l][k] + bih[..]+bhh[..]
// (t' = t fwd, L-1-t bwd).  E = 20 -> 5 K-tiles of 4.
// Guards are branchless: clamped index + 0/1 mask multiply (padded columns
// 30,31 of each gate carry zeros that the scan ignores anyway).
// grid = (L/16, 2 dirs), block = 256 (8 waves, wave = N-tile).
// ---------------------------------------------------------------------------
__global__ __launch_bounds__(256) void gemm_xg1(
    const float* __restrict__ embed, const int* __restrict__ tokens,
    const float* __restrict__ WihF, const float* __restrict__ bihF,
    const float* __restrict__ bhhF, const float* __restrict__ WihB,
    const float* __restrict__ bihB, const float* __restrict__ bhhB,
    float* __restrict__ xgFo, float* __restrict__ xgBo, int L) {
  const int wave = threadIdx.x >> 5;  // N-tile (0..7)
  const int lane = threadIdx.x & 31;
  const int m15 = lane & 15;
  const int kh = lane >> 4;
  const int tb = blockIdx.x * 16;
  const bool back = (blockIdx.y != 0);
  const float* Wih = back ? WihB : WihF;
  const float* bih = back ? bihB : bihF;
  const float* bhh = back ? bhhB : bhhF;
  float* xg = back ? xgBo : xgFo;

  const int q = wave >> 1;                 // gate 0..3 (i,f,g,o)
  const int ln = ((wave & 1) << 4) + m15;  // unit within gate, 0..31
  const float msk = (ln < 30) ? 1.0f : 0.0f;
  const int wrow = (ln < 30) ? (q * 30 + ln) : 0;  // clamped, always in-bounds

  const int tt = tb + m15;
  const int tok = tokens[back ? (L - 1 - tt) : tt];
  const float* erow = embed + tok * 20;
  const float* wr = Wih + wrow * 20;

  v8f acc = {};
#pragma unroll
  for (int kt = 0; kt < 5; ++kt) {
    const int k0 = kt * 4 + kh * 2;
    v2f a, b;
    a.x = erow[k0];
    a.y = erow[k0 + 1];
    b.x = wr[k0] * msk;
    b.y = wr[k0 + 1] * msk;
    acc = wmma_16x16x4(a, b, acc);
  }
  const float bias = (bih[wrow] + bhh[wrow]) * msk;
  const int col = wave * 16 + m15;  // == q*32 + ln
#pragma unroll
  for (int r = 0; r < 8; ++r) {
    const int trow = tb + r + kh * 8;
    xg[(size_t)trow * 128 + col] = acc[r] + bias;
  }
}

// ---------------------------------------------------------------------------
// Layer-2 input projection (both directions). A = h1[L,32] (cols 30,31 zero).
// K = 32 -> 8 K-tiles of 4; weight index clamped at k<30, masked.
// ---------------------------------------------------------------------------
__global__ __launch_bounds__(256) void gemm_xg2(
    const float* __restrict__ h1F, const float* __restrict__ h1B,
    const float* __restrict__ WihF, const float* __restrict__ bihF,
    const float* __restrict__ bhhF, const float* __restrict__ WihB,
    const float* __restrict__ bihB, const float* __restrict__ bhhB,
    float* __restrict__ xgFo, float* __restrict__ xgBo, int L) {
  const int wave = threadIdx.x >> 5;
  const int lane = threadIdx.x & 31;
  const int m15 = lane & 15;
  const int kh = lane >> 4;
  const int tb = blockIdx.x * 16;
  const bool back = (blockIdx.y != 0);
  const float* Wih = back ? WihB : WihF;
  const float* bih = back ? bihB : bihF;
  const float* bhh = back ? bhhB : bhhF;
  const float* h1 = back ? h1B : h1F;
  float* xg = back ? xgBo : xgFo;

  const int q = wave >> 1;
  const int ln = ((wave & 1) << 4) + m15;
  const float nmsk = (ln < 30) ? 1.0f : 0.0f;
  const int wrow = (ln < 30) ? (q * 30 + ln) : 0;

  const float* arow = h1 + (size_t)(tb + m15) * 32;
  const float* wr = Wih + wrow * 30;

  v8f acc = {};
#pragma unroll
  for (int kt = 0; kt < 8; ++kt) {
    const int k0 = kt * 4 + kh * 2;
    const int i0 = (k0 < 30) ? k0 : 0;          // clamp (only kt==7 can hit)
    const int i1 = (k0 + 1 < 30) ? (k0 + 1) : 0;
    const float m0 = (k0 < 30) ? nmsk : 0.0f;
    const float m1 = (k0 + 1 < 30) ? nmsk : 0.0f;
    v2f a, b;
    a.x = arow[k0];
    a.y = arow[k0 + 1];
    b.x = wr[i0] * m0;
    b.y = wr[i1] * m1;
    acc = wmma_16x16x4(a, b, acc);
  }
  const float bias = (bih[wrow] + bhh[wrow]) * nmsk;
  const int col = wave * 16 + m15;
#pragma unroll
  for (int r = 0; r < 8; ++r) {
    const int trow = tb + r + kh * 8;
    xg[(size_t)trow * 128 + col] = acc[r] + bias;
  }
}

// ---------------------------------------------------------------------------
// Persistent single-wave LSTM scan; block 0 = one chain, block 1 = the other
// (fwd/bwd chains independent -> separate WGPs, concurrent). Lane l owns
// hidden unit l; Whh rows pinned in 120 VGPRs/lane (occupancy irrelevant for
// a 2-block latency-bound kernel); h broadcast via LDS double buffer
// (barrier is a no-op for a single-wave workgroup).
// Gate stream: async global->LDS staged chunks of 4 steps (ASYNCcnt) when
// available, else register-software-pipelined global loads one step ahead.
// ---------------------------------------------------------------------------
#define SCH 4  // timesteps per staged chunk (L % SCH == 0)

__global__ __launch_bounds__(32) void lstm_scan2(
    const float* __restrict__ xg0, const float* __restrict__ Whh0,
    float* __restrict__ hout0, const float* __restrict__ xg1,
    const float* __restrict__ Whh1, float* __restrict__ hout1, int L) {
  const float* xg = (blockIdx.x == 0) ? xg0 : xg1;
  const float* Whh = (blockIdx.x == 0) ? Whh0 : Whh1;
  float* hout = (blockIdx.x == 0) ? hout0 : hout1;

  const int l = threadIdx.x;  // hidden unit; lanes 30,31 padded
  const bool act = (l < 30);

  // Recurrent weights pinned in VGPRs: w[q][k] = Whh[q*30+l][k]
  const int wl = act ? l : 0;
  const float wmsk = act ? 1.0f : 0.0f;
  float w[4][30];
#pragma unroll
  for (int q = 0; q < 4; ++q)
#pragma unroll
    for (int k = 0; k < 30; ++k)
      w[q][k] = Whh[(q * 30 + wl) * 30 + k] * wmsk;

  __shared__ float hsh[2][32];
#if USE_ASYNC_STAGE
  __shared__ float xstage[2][SCH * 128];
#endif
  hsh[0][l] = 0.0f;
  hsh[1][l] = 0.0f;
  float c = 0.0f;
  __syncthreads();

  int hb = 0;
  auto step = [&](int t, float gi, float gf, float gg, float go) {
    if (t + 64 < L)  // deep prefetch of the gate stream toward L2/L0
      __builtin_prefetch(&xg[(size_t)(t + 64) * 128 + l], 0, 1);
    float hv[30];
#pragma unroll
    for (int k = 0; k < 30; ++k) hv[k] = hsh[hb][k];
#pragma unroll
    for (int k = 0; k < 30; ++k) {
      gi = fmaf(w[0][k], hv[k], gi);
      gf = fmaf(w[1][k], hv[k], gf);
      gg = fmaf(w[2][k], hv[k], gg);
      go = fmaf(w[3][k], hv[k], go);
    }
    const float ig = fast_sigmoid(gi);
    const float fg = fast_sigmoid(gf);
    const float cg = fast_tanh(gg);
    const float og = fast_sigmoid(go);
    c = fmaf(fg, c, ig * cg);
    float h = og * fast_tanh(c);
    if (!act) h = 0.0f;
    hsh[hb ^ 1][l] = h;
    hout[(size_t)t * 32 + l] = h;
    __syncthreads();
    hb ^= 1;
  };

#if USE_ASYNC_STAGE
  // Each async b128 moves one 128-float row (4 floats/lane); SCH rows per
  // chunk -> ASYNCcnt += SCH per issue batch.
  {
    const float* gs = xg + l * 4;
    float* ls = &xstage[0][l * 4];
#pragma unroll
    for (int r = 0; r < SCH; ++r)
      __builtin_amdgcn_global_load_async_to_lds_b128(
          (as1_v4i*)(uintptr_t)(gs + (size_t)r * 128),
          (as3_v4i*)(uint32_t)(uintptr_t)(ls + r * 128), 0, 0);
  }
  int sb = 0;
  for (int t0 = 0; t0 < L; t0 += SCH) {
    __builtin_amdgcn_s_wait_asynccnt(0);  // chunk t0 resident in LDS
    __syncthreads();
    if (t0 + SCH < L) {  // overlap: fetch next chunk into the other buffer
      const float* gs = xg + (size_t)(t0 + SCH) * 128 + l * 4;
      float* ls = &xstage[sb ^ 1][l * 4];
#pragma unroll
      for (int r = 0; r < SCH; ++r)
        __builtin_amdgcn_global_load_async_to_lds_b128(
            (as1_v4i*)(uintptr_t)(gs + (size_t)r * 128),
            (as3_v4i*)(uint32_t)(uintptr_t)(ls + r * 128), 0, 0);
    }
#pragma unroll
    for (int s = 0; s < SCH; ++s) {
      const float* xr = &xstage[sb][s * 128];
      step(t0 + s, xr[0 * 32 + l], xr[1 * 32 + l], xr[2 * 32 + l],
           xr[3 * 32 + l]);
    }
    sb ^= 1;
  }
#else
  // Fallback: register software pipeline, loads issued one full step early.
  float nx0 = xg[0 * 32 + l], nx1 = xg[1 * 32 + l];
  float nx2 = xg[2 * 32 + l], nx3 = xg[3 * 32 + l];
  for (int t = 0; t < L; ++t) {
    const float gi = nx0, gf = nx1, gg = nx2, go = nx3;
    if (t + 1 < L) {
      const float* xr = xg + (size_t)(t + 1) * 128;
      nx0 = xr[0 * 32 + l];
      nx1 = xr[1 * 32 + l];
      nx2 = xr[2 * 32 + l];
      nx3 = xr[3 * 32 + l];
    }
    step(t, gi, gf, gg, go);
  }
#endif
}

// ---------------------------------------------------------------------------
// Final projection, f and b fused into one WMMA accumulator:
// out[t][v] = hf2[t]·Wlin_f[v] + blin_f[v] + hb2[L-1-t]·Wlin_b[v] + blin_b[v]
// grid = L/16/8 blocks of 8 waves; wave = time-tile.
// ---------------------------------------------------------------------------
__global__ __launch_bounds__(256) void gemm_out(
    const float* __restrict__ hF, const float* __restrict__ hB,
    const float* __restrict__ WlinF, const float* __restrict__ blinF,
    const float* __restrict__ WlinB, const float* __restrict__ blinB,
    float* __restrict__ out, int L) {
  const int wave = threadIdx.x >> 5;
  const int lane = threadIdx.x & 31;
  const int m15 = lane & 15;
  const int kh = lane >> 4;
  const int tb = (blockIdx.x * 8 + wave) * 16;

  const bool nv = (m15 < 10);
  const int n = nv ? m15 : 0;  // clamped vocab column
  const float nmsk = nv ? 1.0f : 0.0f;
  const float* af = hF + (size_t)(tb + m15) * 32;
  const float* ab = hB + (size_t)(L - 1 - (tb + m15)) * 32;
  const float* wf = WlinF + n * 30;
  const float* wb = WlinB + n * 30;

  v8f acc = {};
#pragma unroll
  for (int kt = 0; kt < 8; ++kt) {
    const int k0 = kt * 4 + kh * 2;
    const int i0 = (k0 < 30) ? k0 : 0;
    const int i1 = (k0 + 1 < 30) ? (k0 + 1) : 0;
    const float m0 = (k0 < 30) ? nmsk : 0.0f;
    const float m1 = (k0 + 1 < 30) ? nmsk : 0.0f;
    v2f a, b;
    a.x = af[k0];
    a.y = af[k0 + 1];
    b.x = wf[i0] * m0;
    b.y = wf[i1] * m1;
    acc = wmma_16x16x4(a, b, acc);
    a.x = ab[k0];
    a.y = ab[k0 + 1];
    b.x = wb[i0] * m0;
    b.y = wb[i1] * m1;
    acc = wmma_16x16x4(a, b, acc);
  }
  if (nv) {
    const float bias = blinF[n] + blinB[n];
#pragma unroll
    for (int r = 0; r < 8; ++r) {
      const int trow = tb + r + kh * 8;
      out[(size_t)trow * 10 + n] = acc[r] + bias;
    }
  }
}

// ---------------------------------------------------------------------------
// Host-side launcher.  Workspace (floats), ~201 MB total:
//   xgF [L*128] | xgB [L*128] | hAf [L*32] | hAb [L*32] | hBf [L*32] | hBb [L*32]
// xgF/xgB reused between layer-1 and layer-2 passes.
// ---------------------------------------------------------------------------
extern "C" void kernel_launch(void* const* d_in, const int* in_sizes, int n_in,
                              void* d_out, int out_size, void* d_ws,
                              size_t ws_size, hipStream_t stream) {
  const float* embed = (const float*)d_in[0];
  const float* Wih_f1 = (const float*)d_in[1];
  const float* Whh_f1 = (const float*)d_in[2];
  const float* bih_f1 = (const float*)d_in[3];
  const float* bhh_f1 = (const float*)d_in[4];
  const float* Wih_f2 = (const float*)d_in[5];
  const float* Whh_f2 = (const float*)d_in[6];
  const float* bih_f2 = (const float*)d_in[7];
  const float* bhh_f2 = (const float*)d_in[8];
  const float* Wih_b1 = (const float*)d_in[9];
  const float* Whh_b1 = (const float*)d_in[10];
  const float* bih_b1 = (const float*)d_in[11];
  const float* bhh_b1 = (const float*)d_in[12];
  const float* Wih_b2 = (const float*)d_in[13];
  const float* Whh_b2 = (const float*)d_in[14];
  const float* bih_b2 = (const float*)d_in[15];
  const float* bhh_b2 = (const float*)d_in[16];
  const float* Wlin_f = (const float*)d_in[17];
  const float* blin_f = (const float*)d_in[18];
  const float* Wlin_b = (const float*)d_in[19];
  const float* blin_b = (const float*)d_in[20];
  const int* tokens = (const int*)d_in[21];
  const int L = in_sizes[21];

  float* ws = (float*)d_ws;
  float* xgF = ws;
  float* xgB = xgF + (size_t)L * 128;
  float* hAf = xgB + (size_t)L * 128;
  float* hAb = hAf + (size_t)L * 32;
  float* hBf = hAb + (size_t)L * 32;
  float* hBb = hBf + (size_t)L * 32;

  const dim3 blk(256);
  const dim3 grd(L / 16, 2);

  // 1) layer-1 input projections (WMMA, both dirs, fused embed gather)
  gemm_xg1<<<grd, blk, 0, stream>>>(embed, tokens, Wih_f1, bih_f1, bhh_f1,
                                    Wih_b1, bih_b1, bhh_b1, xgF, xgB, L);
  // 2) layer-1 scans (fwd + bwd chains concurrently)
  lstm_scan2<<<dim3(2), dim3(32), 0, stream>>>(xgF, Whh_f1, hAf, xgB, Whh_b1,
                                               hAb, L);
  // 3) layer-2 input projections (WMMA)
  gemm_xg2<<<grd, blk, 0, stream>>>(hAf, hAb, Wih_f2, bih_f2, bhh_f2, Wih_b2,
                                    bih_b2, bhh_b2, xgF, xgB, L);
  // 4) layer-2 scans
  lstm_scan2<<<dim3(2), dim3(32), 0, stream>>>(xgF, Whh_f2, hBf, xgB, Whh_b2,
                                               hBb, L);
  // 5) fused output projection + direction merge (WMMA)
  gemm_out<<<dim3(L / 16 / 8), blk, 0, stream>>>(
      hBf, hBb, Wlin_f, blin_f, Wlin_b, blin_b, (float*)d_out, L);
}